// PolicyGRPO_25640954757603
// MI455X (gfx1250) — compile-verified
//
#include <hip/hip_runtime.h>
#include <hip/hip_bf16.h>

#define GB 32
#define GT 256
#define GV 32000
#define CHUNK 1024               // floats per async chunk (256 threads * 4)
#define NCHUNK 31                // 31 * 1024 = 31744
#define TAILBASE (NCHUNK * CHUNK)
#define NSTAGE 4                 // async pipeline depth (3 chunks in flight)
#define CLIP_RANGE 0.01f
#define KL_COEFF 0.1f

// ---- CDNA5 async global->LDS copy (ASYNCcnt path, gfx1250) ----
__device__ __forceinline__ void async_copy_b128(unsigned lds_addr, const float* g) {
    asm volatile("global_load_async_to_lds_b128 %0, %1, off"
                 :: "v"(lds_addr), "v"((unsigned long long)g)
                 : "memory");
}
__device__ __forceinline__ void wait_async_le3() {
    asm volatile("s_wait_asynccnt 3" ::: "memory");
}
__device__ __forceinline__ void wait_async_le2() {
    asm volatile("s_wait_asynccnt 2" ::: "memory");
}
__device__ __forceinline__ void wait_async_le1() {
    asm volatile("s_wait_asynccnt 1" ::: "memory");
}
__device__ __forceinline__ void wait_async_le0() {
    asm volatile("s_wait_asynccnt 0" ::: "memory");
}
__device__ __forceinline__ void wait_ds0() {
    asm volatile("s_wait_dscnt 0" ::: "memory");
}

// Online (max, sum-exp) merge: (m1,s1) + (m2,s2)
__device__ __forceinline__ void merge_ms(float& m, float& s, float om, float os) {
    float nm = fmaxf(m, om);
    s = s * __expf(m - nm) + os * __expf(om - nm);
    m = nm;
}

// One block per (b,t) row. Streams GV floats through a 4-stage
// async-to-LDS ring (3 chunks outstanding), computes masked token log-prob.
__global__ __launch_bounds__(256)
void row_logprob_kernel(const float* __restrict__ logits,
                        const int*   __restrict__ output_ids,
                        float*       __restrict__ lp_out) {
    __shared__ float buf[NSTAGE][CHUNK];
    __shared__ float red_m[8];
    __shared__ float red_s[8];

    const int row = blockIdx.x;                  // 0 .. B*T-1
    const int tid = threadIdx.x;                 // 0 .. 255
    const float* src = logits + (size_t)row * GV;

    // Per-lane LDS byte address of stage 0; stage s is + s*CHUNK*4 bytes.
    // Low 32 bits of a generic pointer to LDS == LDS byte offset (aperture form).
    const unsigned lds0 = (unsigned)(unsigned long long)(const void*)&buf[0][tid * 4];

    // Target-token value (thread 0 only) — issued early to overlap.
    int tok = 0;
    float xt = 0.f;
    if (tid == 0) {
        tok = output_ids[row];
        xt = src[tok];
    }

    // Prologue: issue chunks 0..2 into stages 0..2.
    async_copy_b128(lds0,                              src + tid * 4);
    async_copy_b128(lds0 + 1u * CHUNK * sizeof(float), src + 1 * CHUNK + tid * 4);
    async_copy_b128(lds0 + 2u * CHUNK * sizeof(float), src + 2 * CHUNK + tid * 4);

    float m = -INFINITY;
    float s = 0.f;

    #pragma unroll 1
    for (int c = 0; c < NCHUNK; ++c) {
        if (c + 3 < NCHUNK) {
            // Issue chunk c+3 into stage (c+3)&3 (read at iter c-1, already
            // retired by wait_ds0), then retire chunk c (in-order completion).
            async_copy_b128(lds0 + (unsigned)((c + 3) & (NSTAGE - 1)) * (CHUNK * sizeof(float)),
                            src + (size_t)(c + 3) * CHUNK + tid * 4);
            wait_async_le3();
        } else if (c + 2 < NCHUNK) {
            wait_async_le2();
        } else if (c + 1 < NCHUNK) {
            wait_async_le1();
        } else {
            wait_async_le0();
        }
        const float4 x = *(const float4*)&buf[c & (NSTAGE - 1)][tid * 4];
        float cm = fmaxf(fmaxf(x.x, x.y), fmaxf(x.z, x.w));
        float nm = fmaxf(m, cm);
        s = s * __expf(m - nm)
          + __expf(x.x - nm) + __expf(x.y - nm)
          + __expf(x.z - nm) + __expf(x.w - nm);
        m = nm;
        // Ensure my LDS read retired before this stage buffer is overwritten
        // by the async issue of a later iteration (WAR on LDS).
        wait_ds0();
    }

    // Tail: 256 remaining floats, one per thread, direct global load.
    {
        float xr = src[TAILBASE + tid];
        float nm = fmaxf(m, xr);
        s = s * __expf(m - nm) + __expf(xr - nm);
        m = nm;
    }

    // Wave32 reduction of (m, s).
    #pragma unroll
    for (int off = 16; off > 0; off >>= 1) {
        float om = __shfl_xor(m, off, 32);
        float os = __shfl_xor(s, off, 32);
        merge_ms(m, s, om, os);
    }
    if ((tid & 31) == 0) {
        red_m[tid >> 5] = m;
        red_s[tid >> 5] = s;
    }
    __syncthreads();

    if (tid == 0) {
        float M = red_m[0];
        float S = red_s[0];
        #pragma unroll
        for (int w = 1; w < 8; ++w) {
            merge_ms(M, S, red_m[w], red_s[w]);
        }
        // masked token log-prob: x_t - M - log(S); pad token id == 0
        float lp = (tok != 0) ? (xt - M - logf(S)) : 0.f;
        lp_out[row] = lp;
    }
}

// Single block: per-b sums over T for three arrays, then the 32-lane GRPO
// epilogue and final 4 scalars.
__global__ __launch_bounds__(256)
void finalize_kernel(const float* __restrict__ lp,        // [B*T] (masked token lp)
                     const float* __restrict__ old_lp,    // [B*T]
                     const float* __restrict__ pre_lp,    // [B*T]
                     const float* __restrict__ rewards,   // [B]
                     const float* __restrict__ values,    // [B]
                     float*       __restrict__ out) {     // [4]
    __shared__ float s_lp[GB];
    __shared__ float s_old[GB];
    __shared__ float s_pre[GB];

    const int tid  = threadIdx.x;
    const int lane = tid & 31;
    const int wid  = tid >> 5;      // 0..7

    #pragma unroll
    for (int i = 0; i < 4; ++i) {
        const int b = i * 8 + wid;  // covers 0..31
        float a0 = 0.f, a1 = 0.f, a2 = 0.f;
        const int base = b * GT;
        for (int k = lane; k < GT; k += 32) {
            a0 += lp[base + k];
            a1 += old_lp[base + k];
            a2 += pre_lp[base + k];
        }
        #pragma unroll
        for (int off = 16; off > 0; off >>= 1) {
            a0 += __shfl_xor(a0, off, 32);
            a1 += __shfl_xor(a1, off, 32);
            a2 += __shfl_xor(a2, off, 32);
        }
        if (lane == 0) {
            s_lp[b]  = a0;
            s_old[b] = a1;
            s_pre[b] = a2;
        }
    }
    __syncthreads();

    if (tid < GB) {
        const int b = tid;
        const float lps = s_lp[b];
        const float pg_ratio = expf(lps - s_old[b]);
        const float clipped  = fminf(fmaxf(pg_ratio, 1.0f - CLIP_RANGE), 1.0f + CLIP_RANGE);
        const float adv = rewards[b] - values[b];
        const float a = pg_ratio * adv;
        const float c = clipped * adv;
        float per_pg  = -fminf(a, c);
        float per_ind = (c < a) ? 1.0f : 0.0f;
        const float klr = s_pre[b] - lps;
        float per_kl = expf(klr) - klr - 1.0f;

        #pragma unroll
        for (int off = 16; off > 0; off >>= 1) {
            per_pg  += __shfl_xor(per_pg, off, 32);
            per_ind += __shfl_xor(per_ind, off, 32);
            per_kl  += __shfl_xor(per_kl, off, 32);
        }
        if (tid == 0) {
            const float pg_loss  = per_pg  * (1.0f / (float)GB);
            const float kl_term  = KL_COEFF * (per_kl * (1.0f / (float)GB));
            const float clip_rat = per_ind * (1.0f / (float)GB);
            out[0] = pg_loss + kl_term;  // loss
            out[1] = pg_loss;            // pg_loss
            out[2] = kl_term;            // KL_COEFF * kl_loss
            out[3] = clip_rat;           // clip_ratio
        }
    }
}

extern "C" void kernel_launch(void* const* d_in, const int* in_sizes, int n_in,
                              void* d_out, int out_size, void* d_ws, size_t ws_size,
                              hipStream_t stream) {
    const float* logits  = (const float*)d_in[0];   // [B,T,V] f32
    const int*   out_ids = (const int*)  d_in[1];   // [B,T]   i32
    const float* old_lp  = (const float*)d_in[2];   // [B,T]   f32
    const float* pre_lp  = (const float*)d_in[3];   // [B,T]   f32
    const float* rewards = (const float*)d_in[4];   // [B]     f32
    const float* values  = (const float*)d_in[5];   // [B]     f32

    float* lp_ws = (float*)d_ws;                    // B*T floats = 32 KB

    row_logprob_kernel<<<GB * GT, 256, 0, stream>>>(logits, out_ids, lp_ws);
    finalize_kernel<<<1, 256, 0, stream>>>(lp_ws, old_lp, pre_lp, rewards, values,
                                           (float*)d_out);
}